// TargetGuidance_38259568673143
// MI455X (gfx1250) — compile-verified
//
#include <hip/hip_runtime.h>
#include <hip/hip_bf16.h>

typedef __attribute__((ext_vector_type(2))) float v2f;
typedef __attribute__((ext_vector_type(8))) float v8f;

#define WAVES_PER_BLOCK 8

// ---------------------------------------------------------------------------
// Kernel 1: loss_weight = softmax(-||target||) over M targets.
// One block of 512 threads, LDS tree reductions (deterministic).
// ---------------------------------------------------------------------------
__global__ void tg_weights_kernel(const float* __restrict__ target,
                                  float* __restrict__ w, int M) {
    __shared__ float sred[512];
    const int t = threadIdx.x;

    float myv = 0.f;
    float v = -3.0e38f;
    if (t < M) {
        float tx = target[2 * t + 0];
        float ty = target[2 * t + 1];
        myv = -sqrtf(tx * tx + ty * ty);
        v = myv;
    }
    // max-reduce
    sred[t] = v;
    __syncthreads();
    for (int s = blockDim.x / 2; s > 0; s >>= 1) {
        if (t < s) sred[t] = fmaxf(sred[t], sred[t + s]);
        __syncthreads();
    }
    const float mx = sred[0];
    __syncthreads();
    // sum-reduce of exp(v - max)
    const float e = (t < M) ? __expf(myv - mx) : 0.f;
    sred[t] = e;
    __syncthreads();
    for (int s = blockDim.x / 2; s > 0; s >>= 1) {
        if (t < s) sred[t] += sred[t + s];
        __syncthreads();
    }
    const float denom = sred[0];
    if (t < M) w[t] = e / denom;
}

// ---------------------------------------------------------------------------
// Kernel 2: per (batch b, 16-target tile) compute min_n dist via WMMA.
//   dist(n,m) = ||x_n||^2 + ||t_m||^2 - 2 x_n.t_m
// A fragment = -2*x (rows n, K=0..1 used), B fragment = t (cols m),
// C initialized with ||t_m||^2, so D = ||t_m||^2 - 2 x.t per 16x16 tile.
// ||x_n||^2 staged in LDS once per block; running min in 8 accum slots.
// Block = 256 threads = 8 waves; wave w owns N-chunk [w*256, w*256+256).
// ---------------------------------------------------------------------------
__global__ void tg_min_kernel(const float* __restrict__ x,       // (B, N, 4)
                              const float* __restrict__ target,  // (M, 2)
                              const float* __restrict__ w,       // (M,)
                              float* __restrict__ partial,       // (B * M/16,)
                              int N, int M) {
    __shared__ __attribute__((aligned(16))) float s_xn2[2048];
    __shared__ float s_red[WAVES_PER_BLOCK * 16];

    const int b    = blockIdx.x;
    const int m0   = blockIdx.y * 16;
    const int tid  = threadIdx.x;
    const int lane = tid & 31;
    const int wave = tid >> 5;
    const float* __restrict__ xb = x + (size_t)b * (size_t)N * 4;

    // Stage squared norms ||x_n||^2 for this batch into LDS.
    for (int n = tid; n < N; n += blockDim.x) {
        float xx = xb[n * 4 + 0];
        float xy = xb[n * 4 + 1];
        s_xn2[n] = xx * xx + xy * xy;
    }
    __syncthreads();

    // B fragment: 4x16, K rows. Lanes 0-15 hold K=0/1 (t_x, t_y) for col m,
    // lanes 16-31 hold K=2/3 which are zero.
    const int col = lane & 15;
    const int m   = m0 + col;
    const float tx  = target[2 * m + 0];
    const float ty  = target[2 * m + 1];
    const float tm2 = tx * tx + ty * ty;
    v2f bfrag;
    bfrag.x = (lane < 16) ? tx : 0.f;
    bfrag.y = (lane < 16) ? ty : 0.f;

    v8f cinit;
#pragma unroll
    for (int j = 0; j < 8; ++j) cinit[j] = tm2;  // every slot's col == this lane's m

    v8f minv;
#pragma unroll
    for (int j = 0; j < 8; ++j) minv[j] = 3.0e38f;

    const int chunk = N / WAVES_PER_BLOCK;  // 256
    const int nbase = wave * chunk;

    for (int t = 0; t < chunk; t += 16) {
        const int n0 = nbase + t;
        // A fragment: 16x4. Lanes 0-15: row n0+lane, K=0/1 = -2*x; lanes 16-31: K=2/3 = 0.
        const int na = n0 + (lane & 15);
        const float ax = xb[na * 4 + 0];
        const float ay = xb[na * 4 + 1];
        v2f afrag;
        afrag.x = (lane < 16) ? (-2.f * ax) : 0.f;
        afrag.y = (lane < 16) ? (-2.f * ay) : 0.f;

        // D = ||t_m||^2 - 2 x.t  (16 n-rows x 16 m-cols)
        v8f d = __builtin_amdgcn_wmma_f32_16x16x4_f32(
            /*neg_a=*/false, afrag, /*neg_b=*/false, bfrag,
            /*c_mod=*/(short)0, cinit, /*reuse_a=*/false, /*reuse_b=*/false);

        // This lane owns rows n0 + 8*(lane>=16) + j, j=0..7.
        const float4* pq = (const float4*)&s_xn2[n0 + 8 * (lane >> 4)];
        const float4 q0 = pq[0];
        const float4 q1 = pq[1];
        minv[0] = fminf(minv[0], d[0] + q0.x);
        minv[1] = fminf(minv[1], d[1] + q0.y);
        minv[2] = fminf(minv[2], d[2] + q0.z);
        minv[3] = fminf(minv[3], d[3] + q0.w);
        minv[4] = fminf(minv[4], d[4] + q1.x);
        minv[5] = fminf(minv[5], d[5] + q1.y);
        minv[6] = fminf(minv[6], d[6] + q1.z);
        minv[7] = fminf(minv[7], d[7] + q1.w);
    }

    // Min over this lane's 8 rows.
    float mymin = minv[0];
#pragma unroll
    for (int j = 1; j < 8; ++j) mymin = fminf(mymin, minv[j]);
    // Merge the two lane halves (rows 0-7 vs 8-15 of each tile): wave32 shuffle.
    mymin = fminf(mymin, __shfl_xor(mymin, 16, 32));

    if (lane < 16) s_red[wave * 16 + lane] = mymin;
    __syncthreads();

    // Cross-wave min + weighted per-block partial (fixed order, no atomics).
    if (tid < 16) {
        float v = s_red[tid];
#pragma unroll
        for (int wv = 1; wv < WAVES_PER_BLOCK; ++wv)
            v = fminf(v, s_red[wv * 16 + tid]);
        s_red[tid] = w[m0 + tid] * v * (1.0f / (float)M);
    }
    __syncthreads();
    if (tid == 0) {
        float s = 0.f;
#pragma unroll
        for (int i = 0; i < 16; ++i) s += s_red[i];
        partial[blockIdx.y * gridDim.x + blockIdx.x] = s;
    }
}

// ---------------------------------------------------------------------------
// Kernel 3: deterministic final reduction of per-block partials.
// ---------------------------------------------------------------------------
__global__ void tg_reduce_kernel(const float* __restrict__ partial,
                                 float* __restrict__ out, int n) {
    __shared__ float s[256];
    const int t = threadIdx.x;
    float acc = 0.f;
    for (int i = t; i < n; i += 256) acc += partial[i];
    s[t] = acc;
    __syncthreads();
    for (int st = 128; st > 0; st >>= 1) {
        if (t < st) s[t] += s[t + st];
        __syncthreads();
    }
    if (t == 0) out[0] = s[0];
}

extern "C" void kernel_launch(void* const* d_in, const int* in_sizes, int n_in,
                              void* d_out, int out_size, void* d_ws, size_t ws_size,
                              hipStream_t stream) {
    const float* x      = (const float*)d_in[0];  // (B, N, 4) f32
    const float* target = (const float*)d_in[1];  // (M, 2)    f32

    const int N = 2048;
    const int M = in_sizes[1] / 2;            // 512
    const int B = in_sizes[0] / (N * 4);      // 64

    float* wbuf    = (float*)d_ws;            // M floats
    float* partial = wbuf + M;                // B * (M/16) floats

    tg_weights_kernel<<<1, 512, 0, stream>>>(target, wbuf, M);

    dim3 grid(B, M / 16);
    tg_min_kernel<<<grid, 256, 0, stream>>>(x, target, wbuf, partial, N, M);

    tg_reduce_kernel<<<1, 256, 0, stream>>>(partial, (float*)d_out, B * (M / 16));
}